// AbstractRelu_67104569033048
// MI455X (gfx1250) — compile-verified
//
#include <hip/hip_runtime.h>

// DeepPoly ReLU relaxation — pure elementwise, HBM-bandwidth-bound.
// Traffic: 6 streams x 16M x 4B = 402 MB one-touch @ 23.3 TB/s => ~17.3 us floor.
// gfx1250 path: 128-bit coalesced global loads/stores with TH=NT (stream-once,
// don't pollute the 192MB L2), wave32-sized blocks, branchless VALU body
// (v_rcp_f32 instead of IEEE divide, which the compiler otherwise outlines
// into divergent EXEC-mask branches). WMMA/TDM are inapplicable here: no
// matrix structure, no data reuse — direct global<->VGPR is the optimal path.

typedef float float4v __attribute__((ext_vector_type(4)));

__device__ __forceinline__ void deeppoly_relu_elem(float xx, float l, float h,
                                                   float& xo, float& lo, float& ho) {
    const bool crossing = (l < 0.0f) && (h > 0.0f);
    const bool inactive = (h <= 0.0f);

    // crossing => h > 0 and l < 0 => h - l > 0; else denom = 1. Always safe.
    const float safe_denom = crossing ? (h - l) : 1.0f;
    const float inv = __builtin_amdgcn_rcpf(safe_denom);   // v_rcp_f32, ~1 ulp

    // Upper bound: crossing -> chord; inactive -> 0; active -> identity.
    const float ub_slope = crossing ? (h * inv)
                                    : (inactive ? 0.0f : 1.0f);
    const float ub_int   = crossing ? (-(l * h) * inv) : 0.0f;

    // Lower bound slope: crossing -> min-area lambda; inactive -> 0; active -> 1.
    const float lamda    = (l * l > h * h) ? 0.0f : 1.0f;
    const float lb_slope = crossing ? lamda : (inactive ? 0.0f : 1.0f);

    xo = fmaxf(xx, 0.0f);
    ho = ub_slope * h + ub_int;
    lo = lb_slope * l;
}

__global__ __launch_bounds__(256) void deeppoly_relu_vec4(
    const float* __restrict__ x, const float* __restrict__ low,
    const float* __restrict__ high, float* __restrict__ x_out,
    float* __restrict__ low_out, float* __restrict__ high_out, int n4) {
    const int i = blockIdx.x * blockDim.x + threadIdx.x;
    if (i >= n4) return;

    // 128-bit NT loads: global_load_b128 th:TH_LOAD_NT (stream-once).
    const float4v xv = __builtin_nontemporal_load((const float4v*)x    + i);
    const float4v lv = __builtin_nontemporal_load((const float4v*)low  + i);
    const float4v hv = __builtin_nontemporal_load((const float4v*)high + i);

    float4v xo, lo, ho;
#pragma unroll
    for (int k = 0; k < 4; ++k) {
        float a, b, c;
        deeppoly_relu_elem(xv[k], lv[k], hv[k], a, b, c);
        xo[k] = a; lo[k] = b; ho[k] = c;
    }

    // 128-bit NT stores: global_store_b128 th:TH_STORE_NT.
    __builtin_nontemporal_store(xo, (float4v*)x_out    + i);
    __builtin_nontemporal_store(lo, (float4v*)low_out  + i);
    __builtin_nontemporal_store(ho, (float4v*)high_out + i);
}

// Scalar tail for n % 4 != 0 (unused when N is a multiple of 4, e.g. 16M).
__global__ void deeppoly_relu_tail(
    const float* __restrict__ x, const float* __restrict__ low,
    const float* __restrict__ high, float* __restrict__ x_out,
    float* __restrict__ low_out, float* __restrict__ high_out,
    int start, int n) {
    const int i = start + threadIdx.x;
    if (i >= n) return;
    float a, b, c;
    deeppoly_relu_elem(x[i], low[i], high[i], a, b, c);
    x_out[i] = a; low_out[i] = b; high_out[i] = c;
}

extern "C" void kernel_launch(void* const* d_in, const int* in_sizes, int n_in,
                              void* d_out, int out_size, void* d_ws, size_t ws_size,
                              hipStream_t stream) {
    (void)n_in; (void)out_size; (void)d_ws; (void)ws_size;
    const float* x    = (const float*)d_in[0];
    const float* low  = (const float*)d_in[1];
    const float* high = (const float*)d_in[2];
    const int n = in_sizes[0];

    float* out      = (float*)d_out;
    float* x_out    = out;
    float* low_out  = out + (size_t)n;
    float* high_out = out + 2 * (size_t)n;

    const int n4 = n >> 2;                 // float4 elements
    const int threads = 256;               // 8 wave32 waves per block
    const int blocks = (n4 + threads - 1) / threads;
    if (blocks > 0) {
        deeppoly_relu_vec4<<<blocks, threads, 0, stream>>>(
            x, low, high, x_out, low_out, high_out, n4);
    }
    const int rem = n & 3;
    if (rem > 0) {
        deeppoly_relu_tail<<<1, rem, 0, stream>>>(
            x, low, high, x_out, low_out, high_out, n4 << 2, n);
    }
}